// LcRnnCell_90744069030235
// MI455X (gfx1250) — compile-verified
//
#include <hip/hip_runtime.h>
#include <hip/hip_bf16.h>

#define BSZ 8192
#define HD  512
#define ED  512
#define BM  128
#define BN  128
#define BK  16
#define LDSP 18   // row pitch in floats: even (keeps v2f 8B-aligned), bank-spreading

typedef __attribute__((ext_vector_type(2))) float v2f;
typedef __attribute__((ext_vector_type(8))) float v8f;

struct SegPack {
  const float* a[8];
  const float* w[8];
  const float* coef[8];
  int apitch[8];
  int wpitch[8];
  int nseg;
};

// ---------------------------------------------------------------- init flag
__global__ void lcr_init(int* flag) { *flag = 0; }

// ------------------------------------------------- gate dot: one wave per row
__global__ __launch_bounds__(256)
void lcr_rowdot(const float* __restrict__ prev_b, const int* __restrict__ prev_depth,
                const float* __restrict__ w_f, float* __restrict__ fraw, int* flag)
{
  const int wave = threadIdx.x >> 5;
  const int lane = threadIdx.x & 31;
  const int row  = blockIdx.x * 8 + wave;
  const int d    = prev_depth[row];
  const float* bp = prev_b + (size_t)row * (2 * HD) + (size_t)d * HD;
  float s = 0.0f;
  #pragma unroll
  for (int k = lane; k < HD; k += 32) s += bp[k] * w_f[k];
  #pragma unroll
  for (int off = 16; off; off >>= 1) s += __shfl_xor(s, off, 32);
  if (lane == 0) {
    float sg = 1.0f / (1.0f + expf(-s));
    fraw[row] = rintf(sg);                // jnp.round = round-half-even
    if (d != 0) atomicOr(flag, 1);
  }
}

// ------------------------------------------- per-row gating coefficients
__global__ __launch_bounds__(256)
void lcr_finalize(const int* __restrict__ pd, const float* __restrict__ fraw,
                  const int* __restrict__ flag,
                  float* cA0, float* cA1, float* cA2,
                  float* cB0, float* cB1, float* cB2, float* cB3,
                  float* od, float* of, float* oj)
{
  const int i = blockIdx.x * blockDim.x + threadIdx.x;
  const bool all_zero = (*flag == 0);
  const float fr = fraw[i];
  const float f = all_zero ? 1.0f : fr;
  const float j = all_zero ? 0.0f : fr;
  const int nd = pd[i] + (int)(f - j);
  od[i] = (float)nd; of[i] = f; oj[i] = j;
  const float eq = (nd == 1) ? 1.0f : 0.0f;
  const float lt = (nd < 1)  ? 1.0f : 0.0f;
  const float fj = f * j;
  const float nn = (1.0f - f) * (1.0f - j);
  const float fn = f * (1.0f - j);
  cA0[i] = eq * fj + lt;  cA1[i] = eq * nn;  cA2[i] = eq * fn;
  cB0[i] = lt;  cB1[i] = eq * fj;  cB2[i] = eq * nn;  cB3[i] = eq * fn;
}

// ----------------------------------------------- fused multi-segment GEMM
// out[:,1,:] = sum_seg (coef_seg[m] * A_seg) @ W_seg^T  + addcoef[m]*addsrc
// out[:,0,:] = 0
__global__ __launch_bounds__(256)
void lcr_gemm(SegPack sp,
              const float* __restrict__ addsrc, int addpitch,
              const float* __restrict__ addcoef,
              float* __restrict__ outbase)
{
  __shared__ float As[BM][LDSP];
  __shared__ float Bsh[BN][LDSP];

  const int t    = threadIdx.x;
  const int lane = t & 31;
  const int wave = t >> 5;
  const int wm = wave >> 2;     // 0..1 : 64-row slab
  const int wn = wave & 3;      // 0..3 : 32-col slab
  const int h  = lane >> 4;     // lane half
  const int rl = lane & 15;

  const int lrow = t >> 1;          // 0..127 (tile row this thread loads)
  const int lcg  = (t & 1) * 8;     // 0 or 8 (8-float column group)

  const int gm = blockIdx.y * BM + lrow;
  const int gn = blockIdx.x * BN + lrow;

  v8f acc[4][2];
  #pragma unroll
  for (int i = 0; i < 4; ++i)
    #pragma unroll
    for (int jj = 0; jj < 2; ++jj)
      #pragma unroll
      for (int e = 0; e < 8; ++e) acc[i][jj][e] = 0.0f;

  for (int s = 0; s < sp.nseg; ++s) {
    const float cf = sp.coef[s][gm];                       // per-row gate scale
    const float* ab = sp.a[s] + (size_t)gm * sp.apitch[s] + lcg;
    const float* wb = sp.w[s] + (size_t)gn * sp.wpitch[s] + lcg;

    for (int kb = 0; kb < 512; kb += BK) {
      const float4 a0 = *(const float4*)(ab + kb);
      const float4 a1 = *(const float4*)(ab + kb + 4);
      const float4 w0 = *(const float4*)(wb + kb);
      const float4 w1 = *(const float4*)(wb + kb + 4);
      if (kb + BK < 512) {
        __builtin_prefetch(ab + kb + BK, 0, 0);            // global_prefetch_b8
        __builtin_prefetch(wb + kb + BK, 0, 0);
      }
      __syncthreads();                                     // WAR vs prev compute
      As[lrow][lcg + 0] = cf * a0.x;  As[lrow][lcg + 1] = cf * a0.y;
      As[lrow][lcg + 2] = cf * a0.z;  As[lrow][lcg + 3] = cf * a0.w;
      As[lrow][lcg + 4] = cf * a1.x;  As[lrow][lcg + 5] = cf * a1.y;
      As[lrow][lcg + 6] = cf * a1.z;  As[lrow][lcg + 7] = cf * a1.w;
      Bsh[lrow][lcg + 0] = w0.x;  Bsh[lrow][lcg + 1] = w0.y;
      Bsh[lrow][lcg + 2] = w0.z;  Bsh[lrow][lcg + 3] = w0.w;
      Bsh[lrow][lcg + 4] = w1.x;  Bsh[lrow][lcg + 5] = w1.y;
      Bsh[lrow][lcg + 6] = w1.z;  Bsh[lrow][lcg + 7] = w1.w;
      __syncthreads();

      #pragma unroll
      for (int s4 = 0; s4 < 4; ++s4) {
        const int k0 = s4 * 4 + 2 * h;
        v2f af[4], bf[2];
        #pragma unroll
        for (int i = 0; i < 4; ++i)
          af[i] = *(const v2f*)&As[wm * 64 + i * 16 + rl][k0];
        #pragma unroll
        for (int jj = 0; jj < 2; ++jj)
          bf[jj] = *(const v2f*)&Bsh[wn * 32 + jj * 16 + rl][k0];
        #pragma unroll
        for (int i = 0; i < 4; ++i)
          #pragma unroll
          for (int jj = 0; jj < 2; ++jj)
            acc[i][jj] = __builtin_amdgcn_wmma_f32_16x16x4_f32(
                false, af[i], false, bf[jj], (short)0, acc[i][jj], false, false);
      }
    }
  }

  // Epilogue: D layout = VGPR v: lanes0-15 -> M=v, lanes16-31 -> M=v+8; N=rl
  #pragma unroll
  for (int i = 0; i < 4; ++i) {
    const int mb = blockIdx.y * BM + wm * 64 + i * 16 + 8 * h;
    #pragma unroll
    for (int v = 0; v < 8; ++v) {
      const int m = mb + v;
      const float ac = addcoef[m];
      const float* arow = addsrc + (size_t)m * addpitch;
      float* orow = outbase + (size_t)m * (2 * HD);
      #pragma unroll
      for (int jj = 0; jj < 2; ++jj) {
        const int n = blockIdx.x * BN + wn * 32 + jj * 16 + rl;
        orow[HD + n] = acc[i][jj][v] + ac * arow[n];
        orow[n] = 0.0f;                                    // zero plane 0
      }
    }
  }
}

// ----------------------------------------------------------------- launch
extern "C" void kernel_launch(void* const* d_in, const int* in_sizes, int n_in,
                              void* d_out, int out_size, void* d_ws, size_t ws_size,
                              hipStream_t stream)
{
  const float* X    = (const float*)d_in[0];
  const float* pa_  = (const float*)d_in[1];   // prev_a (B,2,H)
  const float* pb_  = (const float*)d_in[2];   // prev_b (B,2,H)
  const int*   pd   = (const int*)d_in[3];
  const float* wf   = (const float*)d_in[4];
  const float* wa00 = (const float*)d_in[5];   // (H, E+H)
  const float* wa10 = (const float*)d_in[6];   // (H, E+H)
  const float* wb00 = (const float*)d_in[7];   // (H, E+2H)
  const float* wb11 = (const float*)d_in[8];   // (H, E+H)
  const float* wb10 = (const float*)d_in[9];   // (H, E+H)
  float* out = (float*)d_out;

  // ws layout (floats): [0] flag(int) | +256 fraw | +256+k*8192 coef arrays
  float* wsf = (float*)d_ws;
  int*   flag = (int*)d_ws;
  float* fraw = wsf + 256;
  float* cA0 = wsf + 256 + 1 * BSZ;
  float* cA1 = wsf + 256 + 2 * BSZ;
  float* cA2 = wsf + 256 + 3 * BSZ;
  float* cB0 = wsf + 256 + 4 * BSZ;
  float* cB1 = wsf + 256 + 5 * BSZ;
  float* cB2 = wsf + 256 + 6 * BSZ;
  float* cB3 = wsf + 256 + 7 * BSZ;

  const size_t NA = (size_t)BSZ * 2 * HD;      // 8,388,608
  float* out_nb = out + NA;
  float* od = out + 2 * NA;
  float* of = od + BSZ;
  float* oj = of + BSZ;

  lcr_init<<<1, 1, 0, stream>>>(flag);
  lcr_rowdot<<<BSZ / 8, 256, 0, stream>>>(pb_, pd, wf, fraw, flag);
  lcr_finalize<<<BSZ / 256, 256, 0, stream>>>(pd, fraw, flag,
                                              cA0, cA1, cA2, cB0, cB1, cB2, cB3,
                                              od, of, oj);

  const float* na1 = out + HD;                 // next_a[:,1,:], pitch 2H
  dim3 grid(HD / BN, BSZ / BM);                // (4, 64)

  // Pass A: next_a1 = cA1*[X|pa]@wa00^T + cA2*[X|pb0]@wa10^T + cA0*pa
  SegPack pA = {};
  pA.a[0] = X;        pA.w[0] = wa00;        pA.coef[0] = cA1; pA.apitch[0] = ED;     pA.wpitch[0] = ED + HD;
  pA.a[1] = pa_ + HD; pA.w[1] = wa00 + ED;   pA.coef[1] = cA1; pA.apitch[1] = 2 * HD; pA.wpitch[1] = ED + HD;
  pA.a[2] = X;        pA.w[2] = wa10;        pA.coef[2] = cA2; pA.apitch[2] = ED;     pA.wpitch[2] = ED + HD;
  pA.a[3] = pb_;      pA.w[3] = wa10 + ED;   pA.coef[3] = cA2; pA.apitch[3] = 2 * HD; pA.wpitch[3] = ED + HD;
  pA.nseg = 4;
  lcr_gemm<<<grid, 256, 0, stream>>>(pA, pa_ + HD, 2 * HD, cA0, out);

  // Pass B: next_b1 = cB1*[X|pb1]@wb11^T + cB2*[X|pa|na1]@wb00^T + cB3*[X|na1]@wb10^T + cB0*pb1
  SegPack pB = {};
  pB.a[0] = X;        pB.w[0] = wb11;             pB.coef[0] = cB1; pB.apitch[0] = ED;     pB.wpitch[0] = ED + HD;
  pB.a[1] = pb_ + HD; pB.w[1] = wb11 + ED;        pB.coef[1] = cB1; pB.apitch[1] = 2 * HD; pB.wpitch[1] = ED + HD;
  pB.a[2] = X;        pB.w[2] = wb00;             pB.coef[2] = cB2; pB.apitch[2] = ED;     pB.wpitch[2] = ED + 2 * HD;
  pB.a[3] = pa_ + HD; pB.w[3] = wb00 + ED;        pB.coef[3] = cB2; pB.apitch[3] = 2 * HD; pB.wpitch[3] = ED + 2 * HD;
  pB.a[4] = na1;      pB.w[4] = wb00 + ED + HD;   pB.coef[4] = cB2; pB.apitch[4] = 2 * HD; pB.wpitch[4] = ED + 2 * HD;
  pB.a[5] = X;        pB.w[5] = wb10;             pB.coef[5] = cB3; pB.apitch[5] = ED;     pB.wpitch[5] = ED + HD;
  pB.a[6] = na1;      pB.w[6] = wb10 + ED;        pB.coef[6] = cB3; pB.apitch[6] = 2 * HD; pB.wpitch[6] = ED + HD;
  pB.nseg = 7;
  lcr_gemm<<<grid, 256, 0, stream>>>(pB, pb_ + HD, 2 * HD, cB0, out_nb);

  (void)in_sizes; (void)n_in; (void)out_size; (void)ws_size;
}